// TensorTrainGaussian_15882789060726
// MI455X (gfx1250) — compile-verified
//
#include <hip/hip_runtime.h>
#include <math.h>

typedef __attribute__((ext_vector_type(2))) float v2f;
typedef __attribute__((ext_vector_type(4))) float v4f;
typedef __attribute__((ext_vector_type(8))) float v8f;

#define MM 32   // features / chain length
#define KK 16   // TT rank

// ---------------------------------------------------------------------------
// Parameter preprocessing (tiny: 8192 elements). Folds log_softmax, softplus,
// log and exp2-rescaling, then stores the three arrays PRE-SWIZZLED into the
// exact per-lane WMMA-B register order used by the hot loop:
//   dest[i][lane][r]  with  lane = k + 16*((j>>1)&1),  r = 2*(j>>2) + (j&1)
// so tt_main fetches one step's params as six global_load_b128.
//   Ap = (logW - log(sigma) - 0.5*log(2*pi)) * log2(e)
//   Sp = sqrt(0.5*log2(e)) / sigma
//   Mp = mu
// ---------------------------------------------------------------------------
__global__ void tt_prep(const float* __restrict__ W_logits,
                        const float* __restrict__ mu,
                        const float* __restrict__ pre_sigma,
                        const float* __restrict__ wk0_logits,
                        float* __restrict__ Ap, float* __restrict__ Sp,
                        float* __restrict__ Mp, float* __restrict__ wk0) {
  int tid = blockIdx.x * blockDim.x + threadIdx.x;
  if (tid < KK) {
    float mx = -1e30f;
    for (int k = 0; k < KK; ++k) mx = fmaxf(mx, wk0_logits[k]);
    float se = 0.f;
    for (int k = 0; k < KK; ++k) se += expf(wk0_logits[k] - mx);
    wk0[tid] = expf(wk0_logits[tid] - mx) / se;
  }
  if (tid >= MM * KK) return;
  int i = tid >> 4, j = tid & 15;
  // log_softmax over axis=1 (the k index) of W_logits[i,k,j]
  const float* wsrc = W_logits + i * 256 + j;
  float mx = -1e30f;
  for (int k = 0; k < KK; ++k) mx = fmaxf(mx, wsrc[k * 16]);
  float se = 0.f;
  for (int k = 0; k < KK; ++k) se += expf(wsrc[k * 16] - mx);
  float lse = mx + logf(se);
  const float LOG2E = 1.4426950408889634f;
  const float HALF_LOG_2PI = 0.9189385332046727f;
  const float SQRT_HALF_LOG2E = 0.8493218002880191f;  // sqrt(0.5*log2(e))
  int lane_h = ((j >> 1) & 1) * 16;
  int r      = 2 * (j >> 2) + (j & 1);
  for (int k = 0; k < KK; ++k) {
    int src = i * 256 + k * 16 + j;
    int dst = i * 256 + (k + lane_h) * 8 + r;  // swizzled [i][lane][r]
    float logW = W_logits[src] - lse;
    float p    = pre_sigma[src];
    float sig  = (p > 20.f) ? p : log1pf(expf(p));  // softplus
    Ap[dst] = (logW - logf(sig) - HALF_LOG_2PI) * LOG2E;
    Sp[dst] = SQRT_HALF_LOG2E / sig;
    Mp[dst] = mu[src];
  }
}

// ---------------------------------------------------------------------------
// Main kernel: one wave32 == TWO samples (params shared, two independent
// WMMA/exp chains for ILP). Chain P <- P * R_i^T via four
// V_WMMA_F32_16X16X4_F32 per sample per step. B (=R^T) is generated straight
// in the WMMA B register layout from the swizzled params; the D-layout ->
// A-layout transpose goes through a wave-private LDS tile (same-wave DS ops
// are in-order). Params for step i+1 are prefetched (register double buffer)
// while step i computes.
// ---------------------------------------------------------------------------
__global__ void __launch_bounds__(256) tt_main(const float* __restrict__ X,
                                               const float* __restrict__ Ap,
                                               const float* __restrict__ Sp,
                                               const float* __restrict__ Mp,
                                               const float* __restrict__ wk0,
                                               float* __restrict__ out, int N) {
  __shared__ float tr[8][512];  // 8 waves x two 16x16 f32 tiles
  const int lane = threadIdx.x & 31;
  const int w    = threadIdx.x >> 5;
  const int n0   = (blockIdx.x * 8 + w) * 2;
  const int n1   = n0 + 1;
  if (n0 >= N) return;
  const int k_idx = lane & 15;  // row (k) this lane owns
  const int half  = lane >> 4;

  // lane i holds x_i of each sample (MM == 32 == wave width)
  float xv0 = X[(size_t)n0 * MM + lane];
  float xv1 = (n1 < N) ? X[(size_t)n1 * MM + lane] : 0.f;

  // P = I in A-operand layout: pa[2t+c] = P[k_idx, 4t + c + 2*half]
  float pa0[8], pa1[8];
#pragma unroll
  for (int r = 0; r < 8; ++r) {
    int j = 4 * (r >> 1) + (r & 1) + 2 * half;
    float v = (k_idx == j) ? 1.0f : 0.0f;
    pa0[r] = v;
    pa1[r] = v;
  }

  const v4f* A4 = (const v4f*)Ap;  // 64 v4f per step
  const v4f* S4 = (const v4f*)Sp;
  const v4f* M4 = (const v4f*)Mp;
  const int lb = lane * 2;

  // preload step 0 params (register double buffer)
  v4f cA0 = A4[lb], cA1 = A4[lb + 1];
  v4f cS0 = S4[lb], cS1 = S4[lb + 1];
  v4f cM0 = M4[lb], cM1 = M4[lb + 1];

  for (int i = 0; i < MM; ++i) {
    // prefetch next step's params while this step computes
    const int nb = ((i < MM - 1) ? i + 1 : i) * 64 + lb;
    v4f nA0 = A4[nb], nA1 = A4[nb + 1];
    v4f nS0 = S4[nb], nS1 = S4[nb + 1];
    v4f nM0 = M4[nb], nM1 = M4[nb + 1];

    // broadcast x_i of both samples into SGPRs
    float x0 = __int_as_float(__builtin_amdgcn_readlane(__float_as_int(xv0), i));
    float x1 = __int_as_float(__builtin_amdgcn_readlane(__float_as_int(xv1), i));

    // B = R^T generated directly in B layout: b[r] = R[k_idx, 4(r>>1)+(r&1)+2h]
    float b0[8], b1[8];
#pragma unroll
    for (int r = 0; r < 8; ++r) {
      float a_ = (r < 4) ? cA0[r] : cA1[r - 4];
      float s_ = (r < 4) ? cS0[r] : cS1[r - 4];
      float m_ = (r < 4) ? cM0[r] : cM1[r - 4];
      float u0 = (x0 - m_) * s_;
      float u1 = (x1 - m_) * s_;
      b0[r] = __builtin_amdgcn_exp2f(__builtin_fmaf(-u0, u0, a_));
      b1[r] = __builtin_amdgcn_exp2f(__builtin_fmaf(-u1, u1, a_));
    }

    // P_new = sum_t A_t(16x4) * B_t(4x16), two independent f32 WMMA chains
    v8f d0 = {0.f, 0.f, 0.f, 0.f, 0.f, 0.f, 0.f, 0.f};
    v8f d1 = {0.f, 0.f, 0.f, 0.f, 0.f, 0.f, 0.f, 0.f};
#pragma unroll
    for (int t = 0; t < 4; ++t) {
      v2f a2, b2;
      a2.x = pa0[2 * t]; a2.y = pa0[2 * t + 1];
      b2.x = b0[2 * t];  b2.y = b0[2 * t + 1];
      d0 = __builtin_amdgcn_wmma_f32_16x16x4_f32(false, a2, false, b2,
                                                 (short)0, d0, false, false);
      a2.x = pa1[2 * t]; a2.y = pa1[2 * t + 1];
      b2.x = b1[2 * t];  b2.y = b1[2 * t + 1];
      d1 = __builtin_amdgcn_wmma_f32_16x16x4_f32(false, a2, false, b2,
                                                 (short)0, d1, false, false);
    }

    // Transpose D layout (row over VGPR/half, col over lane) back to A layout
    float* t0 = &tr[w][0];
    float* t1 = &tr[w][256];
#pragma unroll
    for (int r = 0; r < 8; ++r) {
      t0[(r + 8 * half) * 16 + k_idx] = d0[r];
      t1[(r + 8 * half) * 16 + k_idx] = d1[r];
    }
#pragma unroll
    for (int r = 0; r < 8; ++r) {
      int j = 4 * (r >> 1) + (r & 1) + 2 * half;
      pa0[r] = t0[k_idx * 16 + j];
      pa1[r] = t1[k_idx * 16 + j];
    }

    cA0 = nA0; cA1 = nA1; cS0 = nS0; cS1 = nS1; cM0 = nM0; cM1 = nM1;
  }

  // likelihood = sum_{k,j} wk0[k] * P[k,j]; each lane owns 8 entries of row k
  float s0 = 0.f, s1 = 0.f;
#pragma unroll
  for (int r = 0; r < 8; ++r) { s0 += pa0[r]; s1 += pa1[r]; }
  float wkk = wk0[k_idx];
  float p0 = wkk * s0, p1 = wkk * s1;
#pragma unroll
  for (int off = 16; off > 0; off >>= 1) {
    p0 += __shfl_xor(p0, off, 32);
    p1 += __shfl_xor(p1, off, 32);
  }
  const float LN2 = 0.6931471805599453f;
  const float EPSF = 2.220446049250313e-16f;
  if (lane == 0) {
    out[n0] = LN2 * __builtin_amdgcn_logf(p0 + EPSF);
    if (n1 < N) out[n1] = LN2 * __builtin_amdgcn_logf(p1 + EPSF);
  }
}

// ---------------------------------------------------------------------------
extern "C" void kernel_launch(void* const* d_in, const int* in_sizes, int n_in,
                              void* d_out, int out_size, void* d_ws, size_t ws_size,
                              hipStream_t stream) {
  const float* X          = (const float*)d_in[0];  // (N, 32)
  const float* wk0_logits = (const float*)d_in[1];  // (1, 16)
  const float* W_logits   = (const float*)d_in[2];  // (32, 16, 16)
  const float* mu         = (const float*)d_in[3];  // (32, 16, 16)
  const float* pre_sigma  = (const float*)d_in[4];  // (32, 16, 16)
  float* out = (float*)d_out;

  float* ws  = (float*)d_ws;                 // ~98KB scratch
  float* Ap  = ws;
  float* Sp  = ws + MM * KK * KK;            // +8192 floats
  float* Mp  = ws + 2 * MM * KK * KK;        // +16384
  float* wk0 = ws + 3 * MM * KK * KK;        // +24576

  const int N = in_sizes[0] / MM;

  tt_prep<<<2, 256, 0, stream>>>(W_logits, mu, pre_sigma, wk0_logits,
                                 Ap, Sp, Mp, wk0);
  const int waves_needed = (N + 1) / 2;          // 2 samples per wave
  const int blocks = (waves_needed + 7) / 8;     // 8 waves per block
  tt_main<<<blocks, 256, 0, stream>>>(X, Ap, Sp, Mp, wk0, out, N);
}